// PointNet_SA_Module_13958643712353
// MI455X (gfx1250) — compile-verified
//
#include <hip/hip_runtime.h>
#include <hip/hip_bf16.h>

#define BB_   16
#define NN_   4096
#define SS_   1024
#define KK_   32
#define FIN_  64
#define MTOT  (BB_*SS_*KK_)   /* 524288 columns */
#define NBLK  (MTOT/32)       /* 16384 GEMM blocks */
#define R2_   0.04f
#define EPS_  1e-5f

typedef float v2f __attribute__((ext_vector_type(2)));
typedef float v8f __attribute__((ext_vector_type(8)));

// ---------------------------------------------------------------------------
// Phase 1: farthest point sampling. One block per batch, dist[] in LDS.
// Matches jax scan: emits far (starting at 0) then updates dist and argmaxes.
// Tie-break: lowest index (jnp.argmax semantics).
// ---------------------------------------------------------------------------
__global__ __launch_bounds__(256) void fps_kernel(const float* __restrict__ xyz,
                                                  float* __restrict__ out0,
                                                  float* __restrict__ newxyz) {
  __shared__ float dist[NN_];
  __shared__ float rmax[256];
  __shared__ int   ridx[256];
  const int b = blockIdx.x, t = threadIdx.x;
  const float* xb = xyz + (size_t)b * 3 * NN_;
  for (int i = t; i < NN_; i += 256) dist[i] = 1e10f;
  __syncthreads();
  int far = 0;
  for (int i = 0; i < SS_; i++) {
    const float cx = xb[far], cy = xb[NN_ + far], cz = xb[2*NN_ + far];
    if (t < 3) {
      float v = xb[t*NN_ + far];
      out0[((size_t)b*3 + t)*SS_ + i] = v;          // new_xyz output (B,3,S)
      newxyz[((size_t)b*SS_ + i)*3 + t] = v;        // (B,S,3) scratch
    }
    float best = -1.0f; int besti = 0;
    for (int j = t; j < NN_; j += 256) {
      float dx = xb[j] - cx, dy = xb[NN_ + j] - cy, dz = xb[2*NN_ + j] - cz;
      float d  = dx*dx + dy*dy + dz*dz;
      float nd = fminf(dist[j], d);
      dist[j] = nd;
      if (nd > best) { best = nd; besti = j; }      // strided j ascending -> first-max kept
    }
    rmax[t] = best; ridx[t] = besti;
    __syncthreads();
    for (int off = 128; off > 0; off >>= 1) {
      if (t < off) {
        float ov = rmax[t + off]; int oi = ridx[t + off];
        if (ov > rmax[t] || (ov == rmax[t] && oi < ridx[t])) { rmax[t] = ov; ridx[t] = oi; }
      }
      __syncthreads();
    }
    far = ridx[0];
    __syncthreads();
  }
}

// ---------------------------------------------------------------------------
// Phase 2: ball query. 128 threads = 128 queries per block; xyz tiles in LDS.
// In-order scan reproduces "sorted masked arange, first 32, pad with first".
// ---------------------------------------------------------------------------
#define BQ_TILE 2048
__global__ __launch_bounds__(128) void ball_kernel(const float* __restrict__ xyz,
                                                   const float* __restrict__ newxyz,
                                                   int* __restrict__ idx) {
  __shared__ float lx[BQ_TILE], ly[BQ_TILE], lz[BQ_TILE];
  const int b = blockIdx.x >> 3, t = threadIdx.x;
  const int s = ((blockIdx.x & 7) << 7) + t;
  const float* xb = xyz + (size_t)b * 3 * NN_;
  const int grp = b * SS_ + s;
  const float qx = newxyz[grp*3+0], qy = newxyz[grp*3+1], qz = newxyz[grp*3+2];
  int* op = idx + (size_t)grp * KK_;
  int cnt = 0, first = 0;
  for (int tile = 0; tile < NN_; tile += BQ_TILE) {
    for (int i = t; i < BQ_TILE; i += 128) {
      lx[i] = xb[tile + i]; ly[i] = xb[NN_ + tile + i]; lz[i] = xb[2*NN_ + tile + i];
    }
    __syncthreads();
    if (cnt < KK_) {
      for (int j = 0; j < BQ_TILE; j++) {
        float dx = qx - lx[j], dy = qy - ly[j], dz = qz - lz[j];
        float d2 = dx*dx + dy*dy + dz*dz;
        if (d2 <= R2_) {
          if (cnt == 0) first = tile + j;
          op[cnt] = tile + j;
          if (++cnt >= KK_) break;
        }
      }
    }
    __syncthreads();
  }
  for (int c = cnt; c < KK_; c++) op[c] = first;   // pad with first neighbor
}

// ---------------------------------------------------------------------------
// Fused gather for layer-0 B operand: channel c of x0 at column col.
// c<3: grouped_xyz (gather - centroid); 3..66: grouped points; 67: zero pad.
// ---------------------------------------------------------------------------
__device__ __forceinline__ float load_x0(const float* __restrict__ xyz,
                                         const float* __restrict__ pts,
                                         int bb, int j, float nx, float ny, float nz,
                                         int c) {
  if (c < 3) {
    float v = xyz[((size_t)bb*3 + c)*NN_ + j];
    return v - (c == 0 ? nx : (c == 1 ? ny : nz));
  }
  if (c < 67) return pts[((size_t)bb*FIN_ + (c - 3))*NN_ + j];
  return 0.0f;
}

// Per-wave deterministic per-channel (sum, sumsq) partials for BN stats.
__device__ __forceinline__ void stats_reduce_store(const v8f& acc, int obase, int half,
                                                   int n, int coltile, float* lsum,
                                                   float* lsq) {
  float sr[8], qr[8];
#pragma unroll
  for (int r = 0; r < 8; r++) {
    float sv = acc[r], qv = acc[r]*acc[r];
#pragma unroll
    for (int msk = 1; msk < 16; msk <<= 1) {       // reduce 16 cols of this half
      sv += __shfl_xor(sv, msk, 32);
      qv += __shfl_xor(qv, msk, 32);
    }
    sr[r] = sv; qr[r] = qv;
  }
  if (n == 0 && coltile == 0) {
#pragma unroll
    for (int r = 0; r < 8; r++) { int o = obase + half*8 + r; lsum[o] = sr[r]; lsq[o] = qr[r]; }
  }
  __syncthreads();
  if (n == 0 && coltile == 1) {
#pragma unroll
    for (int r = 0; r < 8; r++) { int o = obase + half*8 + r; lsum[o] += sr[r]; lsq[o] += qr[r]; }
  }
  __syncthreads();
}

// ---------------------------------------------------------------------------
// Layer 0: z1 = W0 (64x67) * x0 (67xM), x0 gathered on the fly. 8 waves/block,
// 4 o-tiles x 2 col-tiles = 64ch x 32cols per block. f32 WMMA 16x16x4.
// ---------------------------------------------------------------------------
__global__ __launch_bounds__(256) void gemm0_kernel(
    const float* __restrict__ W0, const float* __restrict__ xyz,
    const float* __restrict__ pts, const float* __restrict__ newxyz,
    const int* __restrict__ idx, float* __restrict__ Y,
    float* __restrict__ psum, float* __restrict__ psq) {
  __shared__ float lsum[64], lsq[64];
  const int t = threadIdx.x, w = t >> 5, lane = t & 31;
  const int n = lane & 15, half = lane >> 4;
  const int obase = (w & 3) << 4, coltile = w >> 2;
  const int col = (blockIdx.x << 5) + (coltile << 4) + n;
  const int grp = col >> 5;            // b*S + s
  const int bb  = grp >> 10;
  const int j   = idx[col];            // K-invariant gather index
  const float nx = newxyz[grp*3+0], ny = newxyz[grp*3+1], nz = newxyz[grp*3+2];
  const int m = obase + n;
  __builtin_prefetch(&W0[m*67], 0, 3);
  v8f acc = {0.f,0.f,0.f,0.f,0.f,0.f,0.f,0.f};
#pragma unroll
  for (int kb = 0; kb < 68; kb += 4) {
    const int ca = kb + half*2;
    v2f a, bv;
    a.x  = (ca     < 67) ? W0[m*67 + ca]     : 0.0f;
    a.y  = (ca + 1 < 67) ? W0[m*67 + ca + 1] : 0.0f;
    bv.x = load_x0(xyz, pts, bb, j, nx, ny, nz, ca);
    bv.y = load_x0(xyz, pts, bb, j, nx, ny, nz, ca + 1);
    acc = __builtin_amdgcn_wmma_f32_16x16x4_f32(false, a, false, bv, (short)0, acc,
                                                false, false);
  }
#pragma unroll
  for (int r = 0; r < 8; r++) {
    const int o = obase + half*8 + r;
    Y[(size_t)o*MTOT + col] = acc[r];
  }
  stats_reduce_store(acc, obase, half, n, coltile, lsum, lsq);
  if (t < 64) {
    psum[(size_t)t*NBLK + blockIdx.x] = lsum[t];
    psq [(size_t)t*NBLK + blockIdx.x] = lsq[t];
  }
}

// ---------------------------------------------------------------------------
// Layer 1: z2 = W1 (64x64) * relu(scale*z1+shift). Same tiling as layer 0.
// ---------------------------------------------------------------------------
__global__ __launch_bounds__(256) void gemm_mid_kernel(
    const float* __restrict__ Wt, const float* __restrict__ Xin,
    const float* __restrict__ sc, const float* __restrict__ sh,
    float* __restrict__ Y, float* __restrict__ psum, float* __restrict__ psq) {
  __shared__ float lsum[64], lsq[64];
  const int t = threadIdx.x, w = t >> 5, lane = t & 31;
  const int n = lane & 15, half = lane >> 4;
  const int obase = (w & 3) << 4, coltile = w >> 2;
  const int col = (blockIdx.x << 5) + (coltile << 4) + n;
  const int m = obase + n;
  __builtin_prefetch(&Wt[m*64], 0, 3);
  v8f acc = {0.f,0.f,0.f,0.f,0.f,0.f,0.f,0.f};
#pragma unroll
  for (int kb = 0; kb < 64; kb += 4) {
    const int ca = kb + half*2;
    v2f a, bv;
    a.x = Wt[m*64 + ca];
    a.y = Wt[m*64 + ca + 1];
    float x0v = Xin[(size_t)ca*MTOT + col];
    float x1v = Xin[(size_t)(ca + 1)*MTOT + col];
    bv.x = fmaxf(fmaf(x0v, sc[ca],     sh[ca]),     0.0f);
    bv.y = fmaxf(fmaf(x1v, sc[ca + 1], sh[ca + 1]), 0.0f);
    acc = __builtin_amdgcn_wmma_f32_16x16x4_f32(false, a, false, bv, (short)0, acc,
                                                false, false);
  }
#pragma unroll
  for (int r = 0; r < 8; r++) {
    const int o = obase + half*8 + r;
    Y[(size_t)o*MTOT + col] = acc[r];
  }
  stats_reduce_store(acc, obase, half, n, coltile, lsum, lsq);
  if (t < 64) {
    psum[(size_t)t*NBLK + blockIdx.x] = lsum[t];
    psq [(size_t)t*NBLK + blockIdx.x] = lsq[t];
  }
}

// ---------------------------------------------------------------------------
// Layer 2: z3 = W2 (128x64) * relu(scale*z2+shift). 512 thr = 16 waves,
// 8 o-tiles x 2 col-tiles -> 128ch x 32cols = exactly one (b,s) group.
// mode 0: BN partials only (no y3 materialization); mode 1: recompute,
// normalize+relu, max over K=32, write new_points.
// ---------------------------------------------------------------------------
__global__ __launch_bounds__(512) void gemm_last_kernel(
    const float* __restrict__ W2, const float* __restrict__ Xin,
    const float* __restrict__ sc_in, const float* __restrict__ sh_in,
    const float* __restrict__ sc_out, const float* __restrict__ sh_out,
    float* __restrict__ psum, float* __restrict__ psq,
    float* __restrict__ out_np, int mode) {
  __shared__ float lsA[128], lsB[128];
  const int t = threadIdx.x, w = t >> 5, lane = t & 31;
  const int n = lane & 15, half = lane >> 4;
  const int obase = (w & 7) << 4, coltile = w >> 3;
  const int col = (blockIdx.x << 5) + (coltile << 4) + n;
  const int m = obase + n;
  v8f acc = {0.f,0.f,0.f,0.f,0.f,0.f,0.f,0.f};
#pragma unroll
  for (int kb = 0; kb < 64; kb += 4) {
    const int ca = kb + half*2;
    v2f a, bv;
    a.x = W2[m*64 + ca];
    a.y = W2[m*64 + ca + 1];
    float x0v = Xin[(size_t)ca*MTOT + col];
    float x1v = Xin[(size_t)(ca + 1)*MTOT + col];
    bv.x = fmaxf(fmaf(x0v, sc_in[ca],     sh_in[ca]),     0.0f);
    bv.y = fmaxf(fmaf(x1v, sc_in[ca + 1], sh_in[ca + 1]), 0.0f);
    acc = __builtin_amdgcn_wmma_f32_16x16x4_f32(false, a, false, bv, (short)0, acc,
                                                false, false);
  }
  if (mode == 0) {
    float sr[8], qr[8];
#pragma unroll
    for (int r = 0; r < 8; r++) {
      float sv = acc[r], qv = acc[r]*acc[r];
#pragma unroll
      for (int msk = 1; msk < 16; msk <<= 1) {
        sv += __shfl_xor(sv, msk, 32);
        qv += __shfl_xor(qv, msk, 32);
      }
      sr[r] = sv; qr[r] = qv;
    }
    if (n == 0 && coltile == 0) {
#pragma unroll
      for (int r = 0; r < 8; r++) { int o = obase + half*8 + r; lsA[o] = sr[r]; lsB[o] = qr[r]; }
    }
    __syncthreads();
    if (n == 0 && coltile == 1) {
#pragma unroll
      for (int r = 0; r < 8; r++) { int o = obase + half*8 + r; lsA[o] += sr[r]; lsB[o] += qr[r]; }
    }
    __syncthreads();
    if (t < 128) {
      psum[(size_t)t*NBLK + blockIdx.x] = lsA[t];
      psq [(size_t)t*NBLK + blockIdx.x] = lsB[t];
    }
  } else {
    float mr[8];
#pragma unroll
    for (int r = 0; r < 8; r++) {
      const int o = obase + half*8 + r;
      float v = fmaxf(fmaf(acc[r], sc_out[o], sh_out[o]), 0.0f);  // BN + relu
#pragma unroll
      for (int msk = 1; msk < 16; msk <<= 1) v = fmaxf(v, __shfl_xor(v, msk, 32));
      mr[r] = v;                                                  // max over 16 k's
    }
    if (n == 0 && coltile == 0) {
#pragma unroll
      for (int r = 0; r < 8; r++) { int o = obase + half*8 + r; lsA[o] = mr[r]; }
    }
    __syncthreads();
    if (n == 0 && coltile == 1) {
#pragma unroll
      for (int r = 0; r < 8; r++) { int o = obase + half*8 + r; lsA[o] = fmaxf(lsA[o], mr[r]); }
    }
    __syncthreads();
    if (t < 128) {
      const int grp = blockIdx.x, bb = grp >> 10, ss = grp & 1023;
      out_np[((size_t)bb*128 + t)*SS_ + ss] = lsA[t];
    }
  }
}

// ---------------------------------------------------------------------------
// BN stats finalize: per channel c, sum NBLK partials (fixed order -> exactly
// deterministic), emit fused scale/shift (conv bias cancels under BN).
// ---------------------------------------------------------------------------
__global__ __launch_bounds__(256) void bn_reduce_kernel(
    const float* __restrict__ psum, const float* __restrict__ psq,
    const float* __restrict__ g, const float* __restrict__ beta,
    float* __restrict__ scale, float* __restrict__ shift) {
  __shared__ float ls[256], lq[256];
  const int c = blockIdx.x, t = threadIdx.x;
  float s = 0.0f, q = 0.0f;
  for (int i = t; i < NBLK; i += 256) {
    s += psum[(size_t)c*NBLK + i];
    q += psq [(size_t)c*NBLK + i];
  }
  ls[t] = s; lq[t] = q;
  __syncthreads();
  for (int off = 128; off > 0; off >>= 1) {
    if (t < off) { ls[t] += ls[t + off]; lq[t] += lq[t + off]; }
    __syncthreads();
  }
  if (t == 0) {
    const float inv  = 1.0f / (float)MTOT;
    const float mean = ls[0] * inv;
    const float var  = lq[0] * inv - mean*mean;
    const float scl  = g[c] * rsqrtf(var + EPS_);
    scale[c] = scl;
    shift[c] = beta[c] - mean * scl;
  }
}

// ---------------------------------------------------------------------------
extern "C" void kernel_launch(void* const* d_in, const int* in_sizes, int n_in,
                              void* d_out, int out_size, void* d_ws, size_t ws_size,
                              hipStream_t stream) {
  const float* xyz = (const float*)d_in[0];
  const float* pts = (const float*)d_in[1];
  const float* W0  = (const float*)d_in[2];
  const float* g0  = (const float*)d_in[4];
  const float* be0 = (const float*)d_in[5];
  const float* W1  = (const float*)d_in[6];
  const float* g1  = (const float*)d_in[8];
  const float* be1 = (const float*)d_in[9];
  const float* W2  = (const float*)d_in[10];
  const float* g2  = (const float*)d_in[12];
  const float* be2 = (const float*)d_in[13];
  (void)in_sizes; (void)n_in; (void)out_size; (void)ws_size;

  char* p = (char*)d_ws;
  float* newxyz = (float*)p;  p += (size_t)BB_*SS_*3 * sizeof(float);
  int*   idx    = (int*)p;    p += (size_t)MTOT * sizeof(int);
  float* y1     = (float*)p;  p += (size_t)64 * MTOT * sizeof(float);
  float* y2     = (float*)p;  p += (size_t)64 * MTOT * sizeof(float);
  float* psum   = (float*)p;  p += (size_t)128 * NBLK * sizeof(float);
  float* psq    = (float*)p;  p += (size_t)128 * NBLK * sizeof(float);
  float* sc1 = (float*)p; p += 64  * sizeof(float);
  float* sh1 = (float*)p; p += 64  * sizeof(float);
  float* sc2 = (float*)p; p += 64  * sizeof(float);
  float* sh2 = (float*)p; p += 64  * sizeof(float);
  float* sc3 = (float*)p; p += 128 * sizeof(float);
  float* sh3 = (float*)p; p += 128 * sizeof(float);

  float* out0   = (float*)d_out;                 // new_xyz (B,3,S)
  float* out_np = out0 + (size_t)BB_*3*SS_;      // new_points (B,128,S)

  fps_kernel<<<BB_, 256, 0, stream>>>(xyz, out0, newxyz);
  ball_kernel<<<BB_*8, 128, 0, stream>>>(xyz, newxyz, idx);

  gemm0_kernel<<<NBLK, 256, 0, stream>>>(W0, xyz, pts, newxyz, idx, y1, psum, psq);
  bn_reduce_kernel<<<64, 256, 0, stream>>>(psum, psq, g0, be0, sc1, sh1);

  gemm_mid_kernel<<<NBLK, 256, 0, stream>>>(W1, y1, sc1, sh1, y2, psum, psq);
  bn_reduce_kernel<<<64, 256, 0, stream>>>(psum, psq, g1, be1, sc2, sh2);

  gemm_last_kernel<<<NBLK, 512, 0, stream>>>(W2, y2, sc2, sh2, sc3, sh3,
                                             psum, psq, out_np, 0);
  bn_reduce_kernel<<<128, 256, 0, stream>>>(psum, psq, g2, be2, sc3, sh3);
  gemm_last_kernel<<<NBLK, 512, 0, stream>>>(W2, y2, sc2, sh2, sc3, sh3,
                                             psum, psq, out_np, 1);
}